// Solver_40553081209603
// MI455X (gfx1250) — compile-verified
//
#include <hip/hip_runtime.h>
#include <math.h>

typedef __attribute__((ext_vector_type(2))) float v2f;
typedef __attribute__((ext_vector_type(8))) float v8f;

#define Zd 16
#define Nd 466
#define Pd 4
#define Hd 86
#define Td 4096
#define MAX_INNER 8

// =====================================================================
// Kernel A: encoder + strictly-sequential adaptive Dopri5 integration.
// One wave32. A,B live in LDS (padded to kill bank conflicts).
// Lane = (component i = lane&15, j-half = lane>>4). Each g(z) eval:
//   g_i = sum_j z_j * ( A[i][j] + sum_k B[i][j][k] z_k )
// split over the two halves and reduced with shfl_xor(16).
// =====================================================================
__global__ __launch_bounds__(32)
void ode_solver_kernel(const float* __restrict__ n0, const float* __restrict__ p,
                       const float* __restrict__ tstep,
                       const float* __restrict__ Ain, const float* __restrict__ Bin,
                       const float* __restrict__ We1, const float* __restrict__ be1,
                       const float* __restrict__ We2, const float* __restrict__ be2,
                       float* __restrict__ zs) {
  __shared__ __align__(16) float Bs[Zd * 260]; // row stride 260: banks i*4 -> conflict free, 16B aligned
  __shared__ float As[Zd * 17];                // row stride 17: conflict free
  __shared__ float zsh[Zd];
  __shared__ float ysh[Zd];
  __shared__ float hsh[96];

  const int lane = threadIdx.x;
  const int i    = lane & 15;
  const int half = lane >> 4;

  for (int idx = lane; idx < Zd * Zd; idx += 32)
    As[(idx >> 4) * 17 + (idx & 15)] = Ain[idx];
  for (int idx = lane; idx < Zd * Zd * Zd; idx += 32)
    Bs[(idx >> 8) * 260 + (idx & 255)] = Bin[idx];
  __syncthreads();

  // ---------------- encoder ----------------
  for (int j = lane; j < Hd; j += 32) {
    float s = be1[j];
    for (int q = 0; q < Pd; ++q) s = fmaf(p[q],  We1[q * Hd + j], s);
    for (int q = 0; q < Nd; ++q) s = fmaf(n0[q], We1[(Pd + q) * Hd + j], s);
    hsh[j] = tanhf(s);
  }
  __syncthreads();

  float z_i;
  {
    float s = be2[i];
    for (int j = 0; j < Hd; ++j) s = fmaf(hsh[j], We2[j * Zd + i], s);
    z_i = tanhf(s);                 // both halves hold component i
    if (lane < 16) { zsh[i] = z_i; zs[i] = z_i; }  // zs row 0 = z0
  }
  __syncthreads();

  const float* Arow = As + i * 17  + half * 8;
  const float* Brow = Bs + i * 260 + half * 128;

  auto geval = [&](const float* __restrict__ y) -> float {
    float yv[16];
#pragma unroll
    for (int k = 0; k < 16; ++k) yv[k] = y[k];   // broadcast LDS reads
    float acc = 0.f;
#pragma unroll
    for (int jj = 0; jj < 8; ++jj) {
      const float4* bp = reinterpret_cast<const float4*>(Brow + jj * 16);
      float4 b0 = bp[0], b1 = bp[1], b2 = bp[2], b3 = bp[3];
      float m = Arow[jj];
      m = fmaf(b0.x, yv[0],  m); m = fmaf(b0.y, yv[1],  m);
      m = fmaf(b0.z, yv[2],  m); m = fmaf(b0.w, yv[3],  m);
      m = fmaf(b1.x, yv[4],  m); m = fmaf(b1.y, yv[5],  m);
      m = fmaf(b1.z, yv[6],  m); m = fmaf(b1.w, yv[7],  m);
      m = fmaf(b2.x, yv[8],  m); m = fmaf(b2.y, yv[9],  m);
      m = fmaf(b2.z, yv[10], m); m = fmaf(b2.w, yv[11], m);
      m = fmaf(b3.x, yv[12], m); m = fmaf(b3.y, yv[13], m);
      m = fmaf(b3.z, yv[14], m); m = fmaf(b3.w, yv[15], m);
      const float zj = half ? yv[jj + 8] : yv[jj];  // constant indices under unroll
      acc = fmaf(m, zj, acc);
    }
    acc += __shfl_xor(acc, 16, 32);  // combine halves; all lanes hold g_i
    return acc;
  };

  auto fstage = [&](float yi) -> float {
    __syncthreads();                 // single-wave workgroup: lowers to S_NOP
    if (lane < 16) ysh[i] = yi;
    __syncthreads();
    return geval(ysh);
  };

  float t  = tstep[0];
  float dt = 0.01f * (tstep[1] - tstep[0]) + 1e-8f;

  for (int outer = 1; outer < Td; ++outer) {
    const float t_next = tstep[outer];
    for (int it = 0; it < MAX_INNER; ++it) {
      const float remaining = t_next - t;
      const bool  done = (remaining <= 1e-12f);
      const float h = done ? dt : fminf(dt, remaining);

      const float k1 = geval(zsh);
      const float k2 = fstage(z_i + h * (0.2f * k1));
      const float k3 = fstage(z_i + h * ((float)(3.0/40.0) * k1 + (float)(9.0/40.0) * k2));
      const float k4 = fstage(z_i + h * ((float)(44.0/45.0) * k1 + (float)(-56.0/15.0) * k2 +
                                         (float)(32.0/9.0) * k3));
      const float k5 = fstage(z_i + h * ((float)(19372.0/6561.0) * k1 + (float)(-25360.0/2187.0) * k2 +
                                         (float)(64448.0/6561.0) * k3 + (float)(-212.0/729.0) * k4));
      const float k6 = fstage(z_i + h * ((float)(9017.0/3168.0) * k1 + (float)(-355.0/33.0) * k2 +
                                         (float)(46732.0/5247.0) * k3 + (float)(49.0/176.0) * k4 +
                                         (float)(-5103.0/18656.0) * k5));
      const float y5 = z_i + h * ((float)(35.0/384.0) * k1 + (float)(500.0/1113.0) * k3 +
                                  (float)(125.0/192.0) * k4 + (float)(-2187.0/6784.0) * k5 +
                                  (float)(11.0/84.0) * k6);
      const float k7 = fstage(y5);
      const float err = h * ((float)(71.0/57600.0) * k1 + (float)(-71.0/16695.0) * k3 +
                             (float)(71.0/1920.0) * k4 + (float)(-17253.0/339200.0) * k5 +
                             (float)(22.0/525.0) * k6 + (float)(-1.0/40.0) * k7);

      const float scale = 1e-5f + 1e-2f * fmaxf(fabsf(z_i), fabsf(y5));
      const float r = err / scale;
      float s = r * r;                       // each component duplicated in 2 lanes
      s += __shfl_xor(s, 1, 32);
      s += __shfl_xor(s, 2, 32);
      s += __shfl_xor(s, 4, 32);
      s += __shfl_xor(s, 8, 32);
      s += __shfl_xor(s, 16, 32);            // sum32 = 2 * sum_i e2_i
      const float enorm = sqrtf(s * (1.0f / 32.0f));

      const bool accept = (enorm <= 1.0f) && (!done);
      z_i = accept ? y5 : z_i;
      t   = accept ? (t + h) : t;
      const float factor = fminf(10.0f, fmaxf(0.2f, 0.9f * powf(enorm + 1e-10f, -0.2f)));
      dt = done ? dt : dt * factor;

      __syncthreads();
      if (lane < 16) zsh[i] = z_i;
      __syncthreads();
    }
    t = t_next;                               // reference scan sets carry t = t_next
    if (lane < 16) zs[outer * Zd + i] = z_i;
  }
}

// =====================================================================
// Kernel B: decoder  n_s = tanh(zs @ Wd1 + bd1) @ Wd2 + bd2
// One wave per 16-timestep tile; V_WMMA_F32_16X16X4_F32 throughout (fp32
// exact; phase is HBM-store-bound so no precision trade is warranted).
// Fragment striping per CDNA5 ISA:
//   A 16x4 : lanes 0-15 (M=lane, K={4c,4c+1}),  lanes 16-31 (M=lane-16, K={4c+2,4c+3})
//   B 4x16 : lanes 0-15 (N=lane, K={4c,4c+1}),  lanes 16-31 (N=lane-16, K={4c+2,4c+3})
//   C 16x16: VGPR r -> (M = r + 8*(lane>>4), N = lane&15)
// =====================================================================
__global__ __launch_bounds__(32)
void decoder_wmma_kernel(const float* __restrict__ zs,
                         const float* __restrict__ Wd1, const float* __restrict__ bd1,
                         const float* __restrict__ Wd2, const float* __restrict__ bd2,
                         float* __restrict__ out) {
  __shared__ float Hs[16 * 100];   // 16 rows x 96 K-cols padded to stride 100 (bank-friendly)
  const int tile = blockIdx.x;     // 0..255
  const int lane = threadIdx.x;    // 0..31
  const int col  = lane & 15;
  const int half = lane >> 4;
  const int kofs = half * 2;

  // A fragments of the 16x16 z tile (M = time row, K = z dim)
  v2f afr[4];
  const float* zrow = zs + (size_t)(tile * 16 + col) * Zd;
#pragma unroll
  for (int c = 0; c < 4; ++c) {
    afr[c][0] = zrow[4 * c + kofs + 0];
    afr[c][1] = zrow[4 * c + kofs + 1];
  }

  // ---- stage 1: H = tanh(Z @ Wd1 + bd1), N = 86 padded to 6 tiles of 16 ----
#pragma unroll
  for (int n = 0; n < 6; ++n) {
    const int  ncol = n * 16 + col;
    const bool inb  = (ncol < Hd);
    const float bias = inb ? bd1[ncol] : 0.f;
    v8f acc;
#pragma unroll
    for (int r = 0; r < 8; ++r) acc[r] = bias;
#pragma unroll
    for (int c = 0; c < 4; ++c) {
      const int k0 = 4 * c + kofs;
      v2f bfr;
      bfr[0] = inb ? Wd1[(k0 + 0) * Hd + ncol] : 0.f;
      bfr[1] = inb ? Wd1[(k0 + 1) * Hd + ncol] : 0.f;
      acc = __builtin_amdgcn_wmma_f32_16x16x4_f32(false, afr[c], false, bfr,
                                                  (short)0, acc, false, false);
    }
    // pad columns (ncol>=86) got bias=0 and B=0 -> acc==0 -> tanh(0)==0 exactly
#pragma unroll
    for (int r = 0; r < 8; ++r)
      Hs[(r + 8 * half) * 100 + n * 16 + col] = tanhf(acc[r]);
  }
  __syncthreads();

  // ---- stage 2: Out = H @ Wd2 + bd2, K = 86 padded to 88, N = 466 -> 30 tiles ----
  for (int nn = 0; nn < 30; ++nn) {
    const int  ncol = nn * 16 + col;
    const bool inb  = (ncol < Nd);
    const float bias = inb ? bd2[ncol] : 0.f;
    v8f acc;
#pragma unroll
    for (int r = 0; r < 8; ++r) acc[r] = bias;
#pragma unroll
    for (int kk = 0; kk < 22; ++kk) {
      const int k0 = 4 * kk + kofs;
      v2f a2, b2;
      a2[0] = Hs[col * 100 + k0 + 0];          // zero in the K>=86 pad region
      a2[1] = Hs[col * 100 + k0 + 1];
      b2[0] = (inb && (k0 + 0 < Hd)) ? Wd2[(k0 + 0) * Nd + ncol] : 0.f;
      b2[1] = (inb && (k0 + 1 < Hd)) ? Wd2[(k0 + 1) * Nd + ncol] : 0.f;
      acc = __builtin_amdgcn_wmma_f32_16x16x4_f32(false, a2, false, b2,
                                                  (short)0, acc, false, false);
    }
    if (inb) {
#pragma unroll
      for (int r = 0; r < 8; ++r)
        out[(size_t)(tile * 16 + r + 8 * half) * Nd + ncol] = acc[r];
    }
  }
}

extern "C" void kernel_launch(void* const* d_in, const int* in_sizes, int n_in,
                              void* d_out, int out_size, void* d_ws, size_t ws_size,
                              hipStream_t stream) {
  (void)in_sizes; (void)n_in; (void)out_size; (void)ws_size;
  const float* n0  = (const float*)d_in[0];
  const float* p   = (const float*)d_in[1];
  const float* ts  = (const float*)d_in[2];
  const float* A   = (const float*)d_in[3];
  const float* B   = (const float*)d_in[4];
  const float* We1 = (const float*)d_in[5];
  const float* be1 = (const float*)d_in[6];
  const float* We2 = (const float*)d_in[7];
  const float* be2 = (const float*)d_in[8];
  const float* Wd1 = (const float*)d_in[9];
  const float* bd1 = (const float*)d_in[10];
  const float* Wd2 = (const float*)d_in[11];
  const float* bd2 = (const float*)d_in[12];

  float* zs = (float*)d_ws;  // (T,16) f32 = 256 KB latent trajectory

  ode_solver_kernel<<<1, 32, 0, stream>>>(n0, p, ts, A, B, We1, be1, We2, be2, zs);
  decoder_wmma_kernel<<<Td / 16, 32, 0, stream>>>(zs, Wd1, bd1, Wd2, bd2, (float*)d_out);
}